// GATEncoder_88029649698963
// MI455X (gfx1250) — compile-verified
//
#include <hip/hip_runtime.h>
#include <math.h>

#define NEG_SLOPE 0.2f
#define BN_EPS 1e-5f
#define HEADS 4
#define CH 128            // channels per layer output (HEADS*32)
#define N_GRAPHS 64

typedef float v2f __attribute__((ext_vector_type(2)));
typedef float v8f __attribute__((ext_vector_type(8)));

__device__ __forceinline__ float lrelu(float v) { return v > 0.f ? v : NEG_SLOPE * v; }

// ---------------------------------------------------------------------------
// Dense GEMM C[M,128] = A[M,128] * B[128,128] using V_WMMA_F32_16X16X4_F32.
// One block = 256 threads = 8 waves. Block b handles M-tile b (rows 16b..16b+15),
// wave w handles N-tile w (cols 16w..16w+15). K = 128 -> 32 wmma steps of K=4.
// A-frag (16x4): lane l -> M = l%16, VGPR v -> K = 2*(l/16)+v
// B-frag (4x16): lane l -> N = l%16, VGPR v -> K = 2*(l/16)+v
// C/D (16x16):   lane l, VGPR v -> M = v + 8*(l/16), N = l%16
// ---------------------------------------------------------------------------
__global__ void __launch_bounds__(256)
gemm_wmma_f32(const float* __restrict__ A, const float* __restrict__ B,
              float* __restrict__ C) {
    const int wave = threadIdx.x >> 5;
    const int lane = threadIdx.x & 31;
    const int lmod = lane & 15;
    const int lhi  = lane >> 4;              // 0 or 1
    const int mt   = blockIdx.x * 16;
    const int nt   = wave * 16;

    v8f acc = {};
    const float* Arow = A + (size_t)(mt + lmod) * CH;
    #pragma unroll 4
    for (int k0 = 0; k0 < CH; k0 += 4) {
        const int ka = k0 + 2 * lhi;
        v2f a, b;
        a.x = Arow[ka];
        a.y = Arow[ka + 1];
        b.x = B[(size_t)ka * CH + nt + lmod];
        b.y = B[(size_t)(ka + 1) * CH + nt + lmod];
        acc = __builtin_amdgcn_wmma_f32_16x16x4_f32(
            /*neg_a=*/false, a, /*neg_b=*/false, b,
            /*c_mod=*/(short)0, acc, /*reuse_a=*/false, /*reuse_b=*/false);
    }
    #pragma unroll
    for (int v = 0; v < 8; ++v) {
        const int m = mt + v + 8 * lhi;
        C[(size_t)m * CH + nt + lmod] = acc[v];
    }
}

// a_src[n,h] = <h[n,h,:], att_src[h,:]>, same for a_dst; e_max init = self-loop logit
__global__ void node_att(const float* __restrict__ h, const float* __restrict__ att_src,
                         const float* __restrict__ att_dst, float* __restrict__ a_src,
                         float* __restrict__ a_dst, float* __restrict__ e_max, int N) {
    int tid = blockIdx.x * blockDim.x + threadIdx.x;
    if (tid >= N * HEADS) return;
    int n = tid >> 2, hh = tid & 3;
    const float* hr = h + (size_t)n * CH + hh * 32;
    const float* as = att_src + hh * 32;
    const float* ad = att_dst + hh * 32;
    float s = 0.f, d = 0.f;
    #pragma unroll 8
    for (int c = 0; c < 32; ++c) { float v = hr[c]; s += v * as[c]; d += v * ad[c]; }
    a_src[tid] = s;
    a_dst[tid] = d;
    e_max[tid] = lrelu(s + d);   // self-loop edge logit (every segment non-empty)
}

// pass 1: segment max over real edges
__global__ void edge_max(const int* __restrict__ ei, int E, const float* __restrict__ a_src,
                         const float* __restrict__ a_dst, float* e_max) {
    int tid = blockIdx.x * blockDim.x + threadIdx.x;
    if (tid >= E * HEADS) return;
    int e = tid >> 2, hh = tid & 3;
    int s = ei[e], d = ei[E + e];
    float v = lrelu(a_src[s * HEADS + hh] + a_dst[d * HEADS + hh]);
    atomicMax(e_max + d * HEADS + hh, v);
}

// zero agg rows; denom init with self-loop term exp(e_self - m)
__global__ void node_init(const float* __restrict__ a_src, const float* __restrict__ a_dst,
                          const float* __restrict__ e_max, float* __restrict__ denom,
                          float* __restrict__ agg, int N) {
    int tid = blockIdx.x * blockDim.x + threadIdx.x;
    if (tid >= N * CH) return;
    int n = tid >> 7, c = tid & 127;
    agg[tid] = 0.f;
    if (c < HEADS) {
        float e = lrelu(a_src[n * HEADS + c] + a_dst[n * HEADS + c]);
        denom[n * HEADS + c] = __expf(e - e_max[n * HEADS + c]);
    }
}

// pass 2: denom += exp(e - m) over real edges
__global__ void edge_denom(const int* __restrict__ ei, int E, const float* __restrict__ a_src,
                           const float* __restrict__ a_dst, const float* __restrict__ e_max,
                           float* denom) {
    int tid = blockIdx.x * blockDim.x + threadIdx.x;
    if (tid >= E * HEADS) return;
    int e = tid >> 2, hh = tid & 3;
    int s = ei[e], d = ei[E + e];
    float v = lrelu(a_src[s * HEADS + hh] + a_dst[d * HEADS + hh]);
    atomicAdd(denom + d * HEADS + hh, __expf(v - e_max[d * HEADS + hh]));
}

// self-loop message (plain stores, runs before edge_msg in stream order)
__global__ void node_self(const float* __restrict__ a_src, const float* __restrict__ a_dst,
                          const float* __restrict__ e_max, const float* __restrict__ denom,
                          const float* __restrict__ h, float* __restrict__ agg, int N) {
    int tid = blockIdx.x * blockDim.x + threadIdx.x;
    if (tid >= N * CH) return;
    int n = tid >> 7, c = tid & 127, hh = c >> 5;
    float e = lrelu(a_src[n * HEADS + hh] + a_dst[n * HEADS + hh]);
    float alpha = __expf(e - e_max[n * HEADS + hh]) / denom[n * HEADS + hh];
    agg[tid] += alpha * h[tid];
}

// pass 3: one wave (32 lanes) per edge; lane handles 4 channels -> 128 total
__global__ void edge_msg(const int* __restrict__ ei, int E, const float* __restrict__ a_src,
                         const float* __restrict__ a_dst, const float* __restrict__ e_max,
                         const float* __restrict__ denom, const float* __restrict__ h,
                         float* agg) {
    int gid = blockIdx.x * blockDim.x + threadIdx.x;
    int edge = gid >> 5, lane = gid & 31;
    if (edge >= E) return;
    int s = ei[edge], d = ei[E + edge];
    int hh = lane >> 3;                       // head of channels [4*lane, 4*lane+4)
    float e = lrelu(a_src[s * HEADS + hh] + a_dst[d * HEADS + hh]);
    float alpha = __expf(e - e_max[d * HEADS + hh]) / denom[d * HEADS + hh];
    float4 v = ((const float4*)(h + (size_t)s * CH))[lane];
    float* out = agg + (size_t)d * CH + lane * 4;
    atomicAdd(out + 0, alpha * v.x);
    atomicAdd(out + 1, alpha * v.y);
    atomicAdd(out + 2, alpha * v.z);
    atomicAdd(out + 3, alpha * v.w);
}

__global__ void zero_f32(float* p, int n) {
    int tid = blockIdx.x * blockDim.x + threadIdx.x;
    if (tid < n) p[tid] = 0.f;
}

// per-channel sum / sumsq of (agg + bias), block-local then one atomic per block
__global__ void bn_stats(const float* __restrict__ agg, const float* __restrict__ bias,
                         float* sum, float* sumsq, int N) {
    int c = threadIdx.x;                      // blockDim.x == 128
    float b = bias[c], s = 0.f, q = 0.f;
    for (int n = blockIdx.x; n < N; n += gridDim.x) {
        float v = agg[(size_t)n * CH + c] + b;
        s += v; q += v * v;
    }
    atomicAdd(&sum[c], s);
    atomicAdd(&sumsq[c], q);
}

// BN (biased var) + ELU
__global__ void bn_apply(const float* __restrict__ agg, const float* __restrict__ bias,
                         const float* __restrict__ sum, const float* __restrict__ sumsq,
                         const float* __restrict__ gamma, const float* __restrict__ beta,
                         float* __restrict__ out, int N) {
    int tid = blockIdx.x * blockDim.x + threadIdx.x;
    if (tid >= N * CH) return;
    int c = tid & 127;
    float invN = 1.f / (float)N;
    float mu = sum[c] * invN;
    float var = sumsq[c] * invN - mu * mu;
    float v = (agg[tid] + bias[c] - mu) * rsqrtf(var + BN_EPS) * gamma[c] + beta[c];
    out[tid] = v > 0.f ? v : (__expf(v) - 1.f);
}

__global__ void add_bias(float* h2, const float* __restrict__ bias, int N) {
    int tid = blockIdx.x * blockDim.x + threadIdx.x;
    if (tid >= N * CH) return;
    h2[tid] += bias[tid & 127];
}

__global__ void pool_sum(const float* __restrict__ h2, const int* __restrict__ batch,
                         float* pooled, float* counts, int N) {
    int tid = blockIdx.x * blockDim.x + threadIdx.x;
    if (tid >= N * CH) return;
    int n = tid >> 7, c = tid & 127;
    int g = batch[n];
    atomicAdd(&pooled[g * CH + c], h2[tid]);
    if (c == 0) atomicAdd(&counts[g], 1.0f);
}

__global__ void pool_div(const float* __restrict__ pooled, const float* __restrict__ counts,
                         float* __restrict__ out) {
    int tid = blockIdx.x * blockDim.x + threadIdx.x;
    if (tid >= N_GRAPHS * CH) return;
    out[tid] = pooled[tid] / fmaxf(counts[tid >> 7], 1.0f);
}

// ---------------------------------------------------------------------------
extern "C" void kernel_launch(void* const* d_in, const int* in_sizes, int n_in,
                              void* d_out, int out_size, void* d_ws, size_t ws_size,
                              hipStream_t stream) {
    const float* x        = (const float*)d_in[0];
    const int*   ei       = (const int*)d_in[1];
    const int*   batch    = (const int*)d_in[2];
    const float* W1       = (const float*)d_in[3];
    const float* att_src1 = (const float*)d_in[4];
    const float* att_dst1 = (const float*)d_in[5];
    const float* bias1    = (const float*)d_in[6];
    const float* gamma1   = (const float*)d_in[7];
    const float* beta1    = (const float*)d_in[8];
    const float* W2       = (const float*)d_in[9];
    const float* att_src2 = (const float*)d_in[10];
    const float* att_dst2 = (const float*)d_in[11];
    const float* bias2    = (const float*)d_in[12];

    const int N = in_sizes[0] / CH;      // 50000 (multiple of 16)
    const int E = in_sizes[1] / 2;       // 800000

    float* ws = (float*)d_ws;
    const size_t nf = (size_t)N * CH;
    float* hlin   = ws;                  // GEMM output (reused both layers)
    float* agg1   = ws + nf;
    float* hbn    = ws + 2 * nf;
    float* a_src  = ws + 3 * nf;
    float* a_dst  = a_src + (size_t)N * HEADS;
    float* e_max  = a_dst + (size_t)N * HEADS;
    float* denom  = e_max + (size_t)N * HEADS;
    float* bn_sum = denom + (size_t)N * HEADS;   // 128
    float* bn_sq  = bn_sum + CH;                 // 128
    float* pooled = bn_sq + CH;                  // 64*128
    float* counts = pooled + (size_t)N_GRAPHS * CH;  // 64 (contiguous small block)

    float* out_h2 = (float*)d_out;               // layer-2 aggregation target
    float* out_gr = out_h2 + nf;                 // graph representations

    const int B = 256;
    const int gNH  = (N * HEADS + B - 1) / B;
    const int gEH  = (E * HEADS + B - 1) / B;
    const int gNC  = (N * CH + B - 1) / B;
    const int gEW  = (E * 32 + B - 1) / B;

    // zero small scratch (bn_sum..counts are contiguous: 128+128+8192+64)
    zero_f32<<<(8512 + B - 1) / B, B, 0, stream>>>(bn_sum, 2 * CH + N_GRAPHS * CH + N_GRAPHS);

    // ---------------- layer 1 ----------------
    gemm_wmma_f32<<<N / 16, 256, 0, stream>>>(x, W1, hlin);
    node_att<<<gNH, B, 0, stream>>>(hlin, att_src1, att_dst1, a_src, a_dst, e_max, N);
    edge_max<<<gEH, B, 0, stream>>>(ei, E, a_src, a_dst, e_max);
    node_init<<<gNC, B, 0, stream>>>(a_src, a_dst, e_max, denom, agg1, N);
    edge_denom<<<gEH, B, 0, stream>>>(ei, E, a_src, a_dst, e_max, denom);
    node_self<<<gNC, B, 0, stream>>>(a_src, a_dst, e_max, denom, hlin, agg1, N);
    edge_msg<<<gEW, B, 0, stream>>>(ei, E, a_src, a_dst, e_max, denom, hlin, agg1);
    bn_stats<<<512, CH, 0, stream>>>(agg1, bias1, bn_sum, bn_sq, N);
    bn_apply<<<gNC, B, 0, stream>>>(agg1, bias1, bn_sum, bn_sq, gamma1, beta1, hbn, N);

    // ---------------- layer 2 ----------------
    gemm_wmma_f32<<<N / 16, 256, 0, stream>>>(hbn, W2, hlin);
    node_att<<<gNH, B, 0, stream>>>(hlin, att_src2, att_dst2, a_src, a_dst, e_max, N);
    edge_max<<<gEH, B, 0, stream>>>(ei, E, a_src, a_dst, e_max);
    node_init<<<gNC, B, 0, stream>>>(a_src, a_dst, e_max, denom, out_h2, N);
    edge_denom<<<gEH, B, 0, stream>>>(ei, E, a_src, a_dst, e_max, denom);
    node_self<<<gNC, B, 0, stream>>>(a_src, a_dst, e_max, denom, hlin, out_h2, N);
    edge_msg<<<gEW, B, 0, stream>>>(ei, E, a_src, a_dst, e_max, denom, hlin, out_h2);
    add_bias<<<gNC, B, 0, stream>>>(out_h2, bias2, N);

    // ---------------- global mean pool ----------------
    pool_sum<<<gNC, B, 0, stream>>>(out_h2, batch, pooled, counts, N);
    pool_div<<<(N_GRAPHS * CH + B - 1) / B, B, 0, stream>>>(pooled, counts, out_gr);
}